// Banyan_38336878084410
// MI455X (gfx1250) — compile-verified
//
#include <hip/hip_runtime.h>
#include <math.h>

// ---------------------------------------------------------------------------
// Banyan pipeline for MI455X (gfx1250, wave32, WMMA, async-LDS).
//  Phase A (schedule): single-workgroup persistent kernel, 63 serial merge
//    iterations, cosine argmax per row via wave32 dot reductions.
//  Phase B (replay+CE): fused outs@featsT GEMM with V_WMMA_F32_16X16X4_F32,
//    double-buffered B tiles via GLOBAL_LOAD_ASYNC_TO_LDS_B128 (ASYNCcnt),
//    and online softmax (never materializes the TxN logits).
// ---------------------------------------------------------------------------

#define E        256
#define CH       16
#define BB       64
#define LL       64
#define VMAX     30720     // >= 30000, rounded
#define TMAX     8192      // >= 2*63*64 = 8064 targets
#define NODE_CAP 8192      // >= 4095 tokens + 4032 composed + 2 (eos/sos)
#define HASH_CAP 16384
#define NEG_INF  (-__builtin_inff())

typedef float v2f  __attribute__((ext_vector_type(2)));
typedef float v8f  __attribute__((ext_vector_type(8)));
typedef int   b128i __attribute__((vector_size(16)));   // matches builtin param

// ---- CDNA5 async global->LDS (ASYNCcnt-tracked) ---------------------------
__device__ __forceinline__ void async_b128(const void* gsrc, void* ldst) {
#if __has_builtin(__builtin_amdgcn_global_load_async_to_lds_b128)
  __builtin_amdgcn_global_load_async_to_lds_b128(
      (__attribute__((address_space(1))) b128i*)(void*)gsrc,
      (__attribute__((address_space(3))) b128i*)ldst, 0, 0);
#else
  unsigned lo =
      (unsigned)(unsigned long long)(__attribute__((address_space(3))) void*)ldst;
  asm volatile("global_load_async_to_lds_b128 %0, %1, off"
               :: "v"(lo), "v"(gsrc) : "memory");
#endif
}

__device__ __forceinline__ void wait_async0() {
#if __has_builtin(__builtin_amdgcn_s_wait_asynccnt)
  __builtin_amdgcn_s_wait_asynccnt(0);
#else
  asm volatile("s_wait_asynccnt 0" ::: "memory");
#endif
}

__device__ __forceinline__ float waveReduceSum(float v) {
  #pragma unroll
  for (int off = 16; off > 0; off >>= 1) v += __shfl_xor(v, off, 32);
  return v;
}

__device__ __forceinline__ unsigned hashfun(int l, int r) {
  unsigned h = (unsigned)l * 2654435761u;
  h ^= (unsigned)r * 40503u + ((unsigned)r << 16);
  return h & (HASH_CAP - 1);
}

__device__ __forceinline__ int hfind(const int* hL, const int* hR, const int* hV,
                                     int l, int r) {
  unsigned h = hashfun(l, r);
  for (;;) {
    int kl = hL[h];
    if (kl == -1) return -1;
    if (kl == l && hR[h] == r) return hV[h];
    h = (h + 1) & (HASH_CAP - 1);
  }
}

__device__ __forceinline__ void hins(int* hL, int* hR, int* hV,
                                     int l, int r, int v) {
  unsigned h = hashfun(l, r);
  while (hL[h] != -1) h = (h + 1) & (HASH_CAP - 1);
  hL[h] = l; hR[h] = r; hV[h] = v;
}

__device__ __forceinline__ int mapid(int id, int nN) {
  // feats layout: [0..nN) nodes, nN = eos (-2), nN+1 = sos (-1)
  return id >= 0 ? id : (id == -2 ? nN : nN + 1);
}

// --------------------------- K0: init workspace ----------------------------
__global__ void init_kernel(int* present, int* hL, int* counters, float* sig,
                            const float* comp_l, const float* comp_r) {
  int i = blockIdx.x * blockDim.x + threadIdx.x;
  if (i < VMAX)     present[i] = 0;
  if (i < HASH_CAP) hL[i] = -1;
  if (i < 64)       counters[i] = (i == 0) ? -1 : 0;   // counters[0] = maxTok
  if (i < 16)       sig[i]      = 1.f / (1.f + __expf(-comp_l[i]));
  else if (i < 32)  sig[i]      = 1.f / (1.f + __expf(-comp_r[i - 16]));
}

// --------------------------- K1: mark presence -----------------------------
__global__ void mark_kernel(const int* x, int* present, int* counters) {
  int i = blockIdx.x * blockDim.x + threadIdx.x;
  if (i < BB * LL) {
    int t = x[i];
    present[t] = 1;
    atomicMax(&counters[0], t);
  }
}

// ------------- K2: rank scan (unique tokens, ascending order) --------------
__global__ __launch_bounds__(1024)
void scan_kernel(const int* present, int* lookup, int* tokens, int* counters) {
  __shared__ int sc[1024];
  const int tid = threadIdx.x;
  int running = 0;                       // identical in every thread
  for (int base = 0; base < VMAX; base += 1024) {
    int t = base + tid;
    int p = (t < 30000) ? present[t] : 0;
    sc[tid] = p;
    __syncthreads();
    #pragma unroll
    for (int off = 1; off < 1024; off <<= 1) {
      int v = (tid >= off) ? sc[tid - off] : 0;
      __syncthreads();
      sc[tid] += v;
      __syncthreads();
    }
    int incl = sc[tid];
    if (p) {
      int rank = running + incl - 1;
      lookup[t] = rank;
      tokens[rank] = t;
    }
    running += sc[1023];
    __syncthreads();
  }
  if (tid == 0) {
    lookup[counters[0]] = -1;            // max token == padding sentinel
    counters[2] = running - 1;           // nTokens (excludes padding token)
  }
}

// --------------- K3: gather token embeddings + norms -----------------------
__global__ __launch_bounds__(256)
void token_feats_kernel(const float* emb, const int* tokens, const int* counters,
                        float* feats, float* norms) {
  const int lane = threadIdx.x & 31;
  const int gw   = (blockIdx.x * 256 + threadIdx.x) >> 5;
  const int nW   = (gridDim.x * 256) >> 5;
  const int nT   = counters[2];
  for (int row = gw; row < nT; row += nW) {
    const float* src = emb + (size_t)tokens[row] * E;
    float* dst = feats + (size_t)row * E;
    float s = 0.f;
    for (int e = lane; e < E; e += 32) { float v = src[e]; dst[e] = v; s += v * v; }
    s = waveReduceSum(s);
    if (lane == 0) norms[row] = sqrtf(s);
  }
}

// --------------------- K3b: build initial index table ----------------------
__global__ void index_kernel(const int* x, const int* lookup, int* idxInit) {
  int i = blockIdx.x * blockDim.x + threadIdx.x;
  if (i < BB * LL) idxInit[i] = lookup[x[i]];
}

// -------------- K4: merge-tree schedule (single workgroup) -----------------
__global__ __launch_bounds__(1024)
void schedule_kernel(const int* idxInit, float* feats, float* norms,
                     int* hL, int* hR, int* hV, int* g_tgt, int* counters,
                     const float* sig, const float* cb) {
  __shared__ int idxA[BB * LL];
  __shared__ int idxB[BB * LL];
  __shared__ int msRow[BB], actRow[BB], nidRow[BB];
  __shared__ int newL[BB], newR[BB], newId[BB];
  __shared__ int sNewCount;
  const int tid = threadIdx.x, lane = tid & 31, wave = tid >> 5;

  for (int i = tid; i < BB * LL; i += 1024) idxA[i] = idxInit[i];
  int nodeCount = counters[2];           // only tid 0 uses/advances
  int tCount = 0;
  __syncthreads();

  for (int iter = 0; iter < LL - 1; ++iter) {
    const int W = LL - iter;
    int* cur = (iter & 1) ? idxB : idxA;
    int* nxt = (iter & 1) ? idxA : idxB;

    // ---- cosine argmax per row (one wave per row; first-max tie-break) ----
    for (int row = wave; row < BB; row += 32) {
      float best = NEG_INF; int bestp = -1;
      const int* r = cur + row * LL;
      for (int p = 0; p + 1 < W; ++p) {
        int j = r[p + 1];
        if (j < 0) break;                 // padding is suffix-only
        int i0 = r[p];
        const float* fi = feats + (size_t)i0 * E;
        const float* fj = feats + (size_t)j  * E;
        float s = 0.f;
        for (int e = lane; e < E; e += 32) s += fi[e] * fj[e];
        s = waveReduceSum(s);
        float c = s / fmaxf(norms[i0] * norms[j], 1e-8f);
        if (c > best) { best = c; bestp = p; }
      }
      if (lane == 0) { msRow[row] = bestp; actRow[row] = (bestp >= 0); }
    }
    __syncthreads();

    // ---- serial bookkeeping: targets, pair dedup via hash, new-node ids ---
    if (tid == 0) {
      int nc = 0;
      for (int r = 0; r < BB; ++r) {
        if (!actRow[r]) continue;
        int p  = msRow[r];
        int l  = cur[r * LL + p], rr = cur[r * LL + p + 1];
        int ls = (p == 0)      ? -1 : cur[r * LL + p - 1];
        int rs = (p + 2 >= W)  ? -2 : cur[r * LL + p + 2];
        g_tgt[3 * tCount + 0] = l;  g_tgt[3 * tCount + 1] = ls; g_tgt[3 * tCount + 2] = rr;
        g_tgt[3 * tCount + 3] = rr; g_tgt[3 * tCount + 4] = l;  g_tgt[3 * tCount + 5] = rs;
        tCount += 2;
        int nid = hfind(hL, hR, hV, l, rr);
        if (nid < 0) {
          nid = nodeCount++;
          hins(hL, hR, hV, l, rr, nid);
          newL[nc] = l; newR[nc] = rr; newId[nc] = nid; ++nc;
        }
        nidRow[r] = nid;
      }
      sNewCount = nc;
    }
    __syncthreads();

    // ---- compose new node features: out[e] = l[e]*sl[c] + r[e]*sr[c] + cb[c]
    const int nc = sNewCount;
    for (int kk = tid; kk < nc * E; kk += 1024) {
      int ni = kk >> 8, e = kk & (E - 1), c = e & (CH - 1);
      float v = feats[(size_t)newL[ni] * E + e] * sig[c]
              + feats[(size_t)newR[ni] * E + e] * sig[CH + c]
              + cb[c];
      feats[(size_t)newId[ni] * E + e] = v;
    }
    __threadfence();
    __syncthreads();

    // ---- norms of new nodes ----
    for (int ni = wave; ni < nc; ni += 32) {
      const float* f = feats + (size_t)newId[ni] * E;
      float s = 0.f;
      for (int e = lane; e < E; e += 32) { float v = f[e]; s += v * v; }
      s = waveReduceSum(s);
      if (lane == 0) norms[newId[ni]] = sqrtf(s);
    }
    __threadfence();
    __syncthreads();

    // ---- index update + column compaction (one thread per row) ----
    if (tid < BB) {
      int r = tid;
      if (actRow[r]) {
        int p = msRow[r], nid = nidRow[r];
        for (int c = 0; c < p; ++c)       nxt[r * LL + c] = cur[r * LL + c];
        nxt[r * LL + p] = nid;
        for (int c = p + 1; c < W - 1; ++c) nxt[r * LL + c] = cur[r * LL + c + 1];
      } else {
        for (int c = 0; c < W - 1; ++c)   nxt[r * LL + c] = cur[r * LL + c];
      }
    }
    __syncthreads();
  }
  if (tid == 0) { counters[3] = nodeCount; counters[4] = tCount; }
}

// ---------- K5a: append eos/sos rows, zero-pad feats to NODE_CAP -----------
__global__ __launch_bounds__(256)
void pad_kernel(const float* sos, const float* eos, const int* counters,
                float* feats) {
  int row = blockIdx.x, e = threadIdx.x;
  int nN = counters[3];
  if (row == nN)          feats[(size_t)row * E + e] = eos[e];
  else if (row == nN + 1) feats[(size_t)row * E + e] = sos[e];
  else if (row > nN + 1)  feats[(size_t)row * E + e] = 0.f;
}

// --------------------- K5b: build outs rows (contexts) ---------------------
__global__ __launch_bounds__(256)
void outs_kernel(const int* tgt, const int* counters, const float* feats,
                 const float* sig, const float* cb, float* outs) {
  int t = blockIdx.x, e = threadIdx.x, c = e & (CH - 1);
  int T = counters[4], nN = counters[3];
  float v = 0.f;
  if (t < T) {
    int cl = mapid(tgt[3 * t + 1], nN);
    int cr = mapid(tgt[3 * t + 2], nN);
    v = feats[(size_t)cl * E + e] * sig[c]
      + feats[(size_t)cr * E + e] * sig[CH + c]
      + cb[c];
  }
  outs[(size_t)t * E + e] = v;
}

// ------- K6: fused logits GEMM (WMMA f32 16x16x4) + online logsumexp -------
// Block: 64 threads = 2 wave32 waves; rows per block = 32 (16 per wave).
// A (outs, 32x256) async-staged once in LDS; B (feats, 16x256) tiles are
// double-buffered via GLOBAL_LOAD_ASYNC_TO_LDS_B128, overlapping the L2->LDS
// fill of tile ct+1 with the 64-deep WMMA chain on tile ct. Two accumulators
// break the D->C RAW chain for matrix-pipe ILP.
__global__ __launch_bounds__(64)
void ce_gemm_kernel(const float* outs, const float* feats, const int* counters,
                    float* lse) {
  __shared__ float As[32 * E];       // 32 KB
  __shared__ float Bs[2][16 * E];    // 2 x 16 KB
  const int tid  = threadIdx.x;
  const int lane = tid & 31, wave = tid >> 5;
  const int rowBase = blockIdx.x * 32;
  const int Ncols   = counters[3] + 2;
  const int nCT     = (Ncols + 15) >> 4;

  { // prologue: async-stage A (2048 x b128) and B tile 0 (1024 x b128)
    const char* asrc = (const char*)(outs + (size_t)rowBase * E);
    #pragma unroll
    for (int i = 0; i < 32; ++i) {
      int f = i * 64 + tid;
      async_b128(asrc + (size_t)f * 16, (char*)As + (size_t)f * 16);
    }
    const char* bsrc = (const char*)feats;
    #pragma unroll
    for (int i = 0; i < 16; ++i) {
      int f = i * 64 + tid;
      async_b128(bsrc + (size_t)f * 16, (char*)&Bs[0][0] + (size_t)f * 16);
    }
  }

  const int waveRow = wave * 16;
  const int colLane = lane & 15;
  const int kOff    = (lane >> 4) << 1;  // lanes 0-15 -> K+0,K+1 ; 16-31 -> K+2,K+3
  float Mx[8], Sx[8];
  #pragma unroll
  for (int v = 0; v < 8; ++v) { Mx[v] = NEG_INF; Sx[v] = 0.f; }

  for (int ct = 0; ct < nCT; ++ct) {
    wait_async0();        // this wave's issued async tile(s) landed in LDS
    __syncthreads();      // whole tile visible to both waves

    if (ct + 1 < nCT) {   // kick off next B tile into the other buffer
      const char* bsrc = (const char*)(feats + (size_t)(ct + 1) * 16 * E);
      char* bdst = (char*)&Bs[(ct + 1) & 1][0];
      #pragma unroll
      for (int i = 0; i < 16; ++i) {
        int f = i * 64 + tid;
        async_b128(bsrc + (size_t)f * 16, bdst + (size_t)f * 16);
      }
    }

    v8f acc0 = {0.f, 0.f, 0.f, 0.f, 0.f, 0.f, 0.f, 0.f};
    v8f acc1 = {0.f, 0.f, 0.f, 0.f, 0.f, 0.f, 0.f, 0.f};
    const float* aRow = &As[(waveRow + colLane) * E + kOff];
    const float* bRow = &Bs[ct & 1][colLane * E + kOff];
    #pragma unroll 4
    for (int k = 0; k < E; k += 8) {
      v2f a0 = *(const v2f*)(aRow + k);
      v2f b0 = *(const v2f*)(bRow + k);
      acc0 = __builtin_amdgcn_wmma_f32_16x16x4_f32(false, a0, false, b0,
                                                   (short)0, acc0, false, false);
      v2f a1 = *(const v2f*)(aRow + k + 4);
      v2f b1 = *(const v2f*)(bRow + k + 4);
      acc1 = __builtin_amdgcn_wmma_f32_16x16x4_f32(false, a1, false, b1,
                                                   (short)0, acc1, false, false);
    }

    const int col = (ct << 4) + colLane;
    if (col < Ncols) {
      #pragma unroll
      for (int v = 0; v < 8; ++v) {     // online softmax update
        float x  = acc0[v] + acc1[v];
        float m2 = fmaxf(Mx[v], x);
        Sx[v] = Sx[v] * __expf(Mx[v] - m2) + __expf(x - m2);
        Mx[v] = m2;
      }
    }
  }

  // combine (max,sum) across the 16 lanes holding each row's column subsets
  #pragma unroll
  for (int v = 0; v < 8; ++v) {
    #pragma unroll
    for (int off = 1; off < 16; off <<= 1) {
      float mo = __shfl_xor(Mx[v], off, 32);
      float so = __shfl_xor(Sx[v], off, 32);
      float m2 = fmaxf(Mx[v], mo);
      Sx[v] = Sx[v] * __expf(Mx[v] - m2) + so * __expf(mo - m2);
      Mx[v] = m2;
    }
  }
  if (colLane == 0) {
    int half = lane >> 4;                // lanes<16: rows 0-7 ; lanes>=16: rows 8-15
    #pragma unroll
    for (int v = 0; v < 8; ++v)
      lse[rowBase + waveRow + half * 8 + v] = Mx[v] + __logf(Sx[v]);
  }
}

// ------------------- K7: logit at the target column ------------------------
__global__ __launch_bounds__(256)
void tlogit_kernel(const int* tgt, const int* counters, const float* outs,
                   const float* feats, float* tlog) {
  const int lane = threadIdx.x & 31;
  const int t = blockIdx.x * 8 + (threadIdx.x >> 5);
  const int T = counters[4], nN = counters[3];
  if (t >= T) return;
  const float* o = outs  + (size_t)t * E;
  const float* f = feats + (size_t)mapid(tgt[3 * t], nN) * E;
  float s = 0.f;
  for (int e = lane; e < E; e += 32) s += o[e] * f[e];
  s = waveReduceSum(s);
  if (lane == 0) tlog[t] = s;
}

// ----------------------- K8: final mean NLL reduction ----------------------
__global__ __launch_bounds__(1024)
void loss_kernel(const float* lse, const float* tlog, const int* counters,
                 float* out) {
  __shared__ float sm[1024];
  const int tid = threadIdx.x;
  const int T = counters[4];
  float s = 0.f;
  for (int t = tid; t < T; t += 1024) s += lse[t] - tlog[t];
  sm[tid] = s;
  __syncthreads();
  for (int off = 512; off > 0; off >>= 1) {
    if (tid < off) sm[tid] += sm[tid + off];
    __syncthreads();
  }
  if (tid == 0) out[0] = sm[0] / (float)T;
}

// ---------------------------------------------------------------------------
extern "C" void kernel_launch(void* const* d_in, const int* in_sizes, int n_in,
                              void* d_out, int out_size, void* d_ws,
                              size_t ws_size, hipStream_t stream) {
  (void)in_sizes; (void)n_in; (void)out_size; (void)ws_size;
  const int*   x      = (const int*)  d_in[0];
  const float* emb    = (const float*)d_in[1];
  const float* comp_l = (const float*)d_in[2];
  const float* comp_r = (const float*)d_in[3];
  const float* cb     = (const float*)d_in[4];
  const float* sos    = (const float*)d_in[5];
  const float* eos    = (const float*)d_in[6];

  char* ws = (char*)d_ws;
  size_t off = 0;
  auto alloc = [&](size_t bytes) -> char* {
    char* p = ws + off;
    off = (off + bytes + 255) & ~(size_t)255;
    return p;
  };
  int*   counters = (int*)  alloc(64 * 4);
  float* sig      = (float*)alloc(32 * 4);
  int*   present  = (int*)  alloc((size_t)VMAX * 4);
  int*   lookup   = (int*)  alloc((size_t)VMAX * 4);
  int*   tokens   = (int*)  alloc((size_t)NODE_CAP * 4);
  int*   idxInit  = (int*)  alloc((size_t)BB * LL * 4);
  int*   hL       = (int*)  alloc((size_t)HASH_CAP * 4);
  int*   hR       = (int*)  alloc((size_t)HASH_CAP * 4);
  int*   hV       = (int*)  alloc((size_t)HASH_CAP * 4);
  int*   tgt      = (int*)  alloc((size_t)3 * TMAX * 4);
  float* lse      = (float*)alloc((size_t)TMAX * 4);
  float* tlog     = (float*)alloc((size_t)TMAX * 4);
  float* norms    = (float*)alloc((size_t)NODE_CAP * 4);
  float* feats    = (float*)alloc((size_t)NODE_CAP * E * 4);   // 8 MB
  float* outs     = (float*)alloc((size_t)TMAX * E * 4);       // 8 MB

  init_kernel<<<(VMAX + 255) / 256, 256, 0, stream>>>(present, hL, counters,
                                                      sig, comp_l, comp_r);
  mark_kernel<<<(BB * LL + 255) / 256, 256, 0, stream>>>(x, present, counters);
  scan_kernel<<<1, 1024, 0, stream>>>(present, lookup, tokens, counters);
  token_feats_kernel<<<64, 256, 0, stream>>>(emb, tokens, counters, feats, norms);
  index_kernel<<<(BB * LL + 255) / 256, 256, 0, stream>>>(x, lookup, idxInit);
  schedule_kernel<<<1, 1024, 0, stream>>>(idxInit, feats, norms, hL, hR, hV,
                                          tgt, counters, sig, cb);
  pad_kernel<<<NODE_CAP, 256, 0, stream>>>(sos, eos, counters, feats);
  outs_kernel<<<TMAX, 256, 0, stream>>>(tgt, counters, feats, sig, cb, outs);
  ce_gemm_kernel<<<TMAX / 32, 64, 0, stream>>>(outs, feats, counters, lse);
  tlogit_kernel<<<TMAX / 8, 256, 0, stream>>>(tgt, counters, outs, feats, tlog);
  loss_kernel<<<1, 1024, 0, stream>>>(lse, tlog, counters, (float*)d_out);
}